// Attention2_29557964931671
// MI455X (gfx1250) — compile-verified
//
#include <hip/hip_runtime.h>
#include <hip/hip_bf16.h>

// Transformer-XL relative attention for MI455X (gfx1250), fp32 WMMA path.
//
// Roofline: p_attn output is 536 MB => the kernel is bandwidth bound
// (~1.6 GB total traffic with the 2-pass score materialization ~= 70us at
// 23.3 TB/s). Compute is ~52 GFLOP fp32 done with V_WMMA_F32_16X16X4_F32,
// the only full-precision matrix instruction on CDNA5. Pass-1 uses two
// independent WMMA accumulator chains (AC and BD) interleaved to halve the
// D->C RAW dependency depth on the matrix pipe.

typedef float v2f __attribute__((ext_vector_type(2)));
typedef float v8f __attribute__((ext_vector_type(8)));

#define Bn 2
#define Sn 2048
#define Hn 16
#define Dn 64
#define NTILES (Sn / 16)   // 128 k-tiles per row

// LDS P-transpose scratch: 4 waves x 16 rows x stride 20 dwords (pad => no bank conflicts)
#define PSTRIDE 20

__global__ __launch_bounds__(128) void txl_attn_kernel(
    const float* __restrict__ qg, const float* __restrict__ kg,
    const float* __restrict__ vg, const float* __restrict__ rg,
    const float* __restrict__ rbias, const float* __restrict__ rwbias,
    float* __restrict__ outg, float* __restrict__ pattn)
{
    __shared__ float smem[4][16 * PSTRIDE];

    const int lane = threadIdx.x & 31;
    const int wave = threadIdx.x >> 5;
    const int n    = lane & 15;   // column / N index (also A-row index m)
    const int hh   = lane >> 4;   // half-wave select (K-pair / row-group select)

    // blockIdx.x = ((b*H + h) * 32) + qb   ; qb in [0,32): 64-row q blocks
    const int qb = blockIdx.x & 31;
    const int bh = blockIdx.x >> 5;       // b*H + h
    const int h  = bh & (Hn - 1);
    const int b  = bh >> 4;

    const int qt = qb * 4 + wave;         // this wave's 16-row q tile index
    const int q0 = qt * 16;

    // ---------------- A-matrix fragments: [q + r_w_bias | q + r_bias], 16x128 ----------------
    // f32 A layout (16x4 per wmma): lanes 0-15 hold K=4t,4t+1 ; lanes 16-31 hold K=4t+2,4t+3
    v2f arw[16], arr[16];
    {
        const float* qrow = qg + (((b * Sn + (q0 + n)) * Hn + h) * Dn);
        const float* wb   = rwbias + h * Dn;
        const float* bb   = rbias  + h * Dn;
        #pragma unroll
        for (int t = 0; t < 16; ++t) {
            const int d0 = 4 * t + 2 * hh;
            v2f qv = *(const v2f*)(qrow + d0);
            v2f wv = *(const v2f*)(wb + d0);
            v2f bv = *(const v2f*)(bb + d0);
            arw[t] = qv + wv;
            arr[t] = qv + bv;
        }
    }

    // ---------------- Pass 1: scores -> p_attn (raw), online row max & sum ----------------
    float mrow[8], lrow[8];
    #pragma unroll
    for (int j = 0; j < 8; ++j) { mrow[j] = -3.0e38f; lrow[j] = 0.0f; }

    for (int kt = 0; kt <= qt; ++kt) {
        const int k0 = kt * 16;
        const float* krow = kg + (((b * Sn + (k0 + n)) * Hn + h) * Dn);
        const float* rrow = rg + (((b * Sn + (k0 + n)) * Hn + h) * Dn);

        // Two independent accumulator chains (AC and BD), interleaved.
        v8f c0 = {};
        v8f c1 = {};
        #pragma unroll
        for (int t = 0; t < 16; ++t) {
            v2f bk = *(const v2f*)(krow + 4 * t + 2 * hh);
            v2f br = *(const v2f*)(rrow + 4 * t + 2 * hh);
            c0 = __builtin_amdgcn_wmma_f32_16x16x4_f32(false, arw[t], false, bk,
                                                       (short)0, c0, false, false);
            c1 = __builtin_amdgcn_wmma_f32_16x16x4_f32(false, arr[t], false, br,
                                                       (short)0, c1, false, false);
        }
        v8f c = c0 + c1;

        const bool diag = (kt == qt);
        #pragma unroll
        for (int j = 0; j < 8; ++j) {
            const int row = q0 + j + 8 * hh;
            float s = c[j] * 0.125f;                       // 1/sqrt(64)
            if (diag && (k0 + n) > row) s = -1.0e9f;       // causal mask
            pattn[(bh * Sn + row) * Sn + k0 + n] = s;      // raw score

            // row reduction across the 16 lanes of this half-wave
            float tm = s;
            tm = fmaxf(tm, __shfl_xor(tm, 1));
            tm = fmaxf(tm, __shfl_xor(tm, 2));
            tm = fmaxf(tm, __shfl_xor(tm, 4));
            tm = fmaxf(tm, __shfl_xor(tm, 8));
            const float mn = fmaxf(mrow[j], tm);
            float e = __expf(s - mn);
            e += __shfl_xor(e, 1);
            e += __shfl_xor(e, 2);
            e += __shfl_xor(e, 4);
            e += __shfl_xor(e, 8);
            lrow[j] = lrow[j] * __expf(mrow[j] - mn) + e;
            mrow[j] = mn;
        }
    }

    // zero-fill fully masked tiles (softmax of -1e9 == 0; d_out is poisoned)
    for (int kt = qt + 1; kt < NTILES; ++kt) {
        #pragma unroll
        for (int j = 0; j < 8; ++j) {
            pattn[(bh * Sn + (q0 + j + 8 * hh)) * Sn + kt * 16 + n] = 0.0f;
        }
    }

    float invl[8];
    #pragma unroll
    for (int j = 0; j < 8; ++j) invl[j] = 1.0f / lrow[j];

    // ---------------- Pass 2: normalize probabilities, O += P @ V ----------------
    v8f o[4];
    #pragma unroll
    for (int db = 0; db < 4; ++db) o[db] = (v8f){};

    for (int kt = 0; kt <= qt; ++kt) {
        const int k0 = kt * 16;

        // load raw scores (C layout, coalesced), normalize, write final p, stage in LDS
        #pragma unroll
        for (int j = 0; j < 8; ++j) {
            const int row = q0 + j + 8 * hh;
            const int idx = (bh * Sn + row) * Sn + k0 + n;
            const float s = pattn[idx];
            const float p = __expf(s - mrow[j]) * invl[j];
            pattn[idx] = p;
            smem[wave][(j + 8 * hh) * PSTRIDE + n] = p;
        }
        asm volatile("s_wait_dscnt 0x0" ::: "memory");   // intra-wave LDS RAW

        // read P back transposed into A-matrix layout (ds_load_b64, conflict-free)
        v2f pa[4];
        #pragma unroll
        for (int t = 0; t < 4; ++t) {
            pa[t] = *(const v2f*)&smem[wave][n * PSTRIDE + 4 * t + 2 * hh];
        }

        // O[16q x 64d] += P[16q x 16k] * V[16k x 64d]
        // 4 independent accumulator chains (d-blocks) x 4 K-chunks.
        const float* vb = vg + (((size_t)(b * Sn + k0) * Hn + h) * Dn);
        #pragma unroll
        for (int t = 0; t < 4; ++t) {
            const int kk = 4 * t + 2 * hh;
            const float* v0p = vb + (size_t)kk * (Hn * Dn);
            const float* v1p = v0p + (Hn * Dn);
            #pragma unroll
            for (int db = 0; db < 4; ++db) {
                v2f bf;
                bf.x = v0p[db * 16 + n];
                bf.y = v1p[db * 16 + n];
                o[db] = __builtin_amdgcn_wmma_f32_16x16x4_f32(false, pa[t], false, bf,
                                                              (short)0, o[db], false, false);
            }
        }
    }

    // ---------------- store O: out[b,h,s,d] ----------------
    #pragma unroll
    for (int db = 0; db < 4; ++db) {
        #pragma unroll
        for (int j = 0; j < 8; ++j) {
            const int row = q0 + j + 8 * hh;
            outg[(bh * Sn + row) * Dn + db * 16 + n] = o[db][j];
        }
    }
}

extern "C" void kernel_launch(void* const* d_in, const int* in_sizes, int n_in,
                              void* d_out, int out_size, void* d_ws, size_t ws_size,
                              hipStream_t stream) {
    const float* q   = (const float*)d_in[0];
    const float* k   = (const float*)d_in[1];
    const float* v   = (const float*)d_in[2];
    const float* r   = (const float*)d_in[3];
    const float* rb  = (const float*)d_in[4];   // r_bias
    const float* rwb = (const float*)d_in[5];   // r_w_bias
    // d_in[6] = causal mask (tril) -- computed analytically in-kernel.

    float* out   = (float*)d_out;                          // [B,H,S,D]
    float* pattn = out + (size_t)Bn * Hn * Sn * Dn;        // [B,H,S,S]

    dim3 grid(Bn * Hn * (Sn / 64));
    dim3 block(128);   // 4 wave32 waves, one 16-row q tile each
    txl_attn_kernel<<<grid, block, 0, stream>>>(q, k, v, r, rb, rwb, out, pattn);
}